// SCANCircuitV4C_27144193310733
// MI455X (gfx1250) — compile-verified
//
#include <hip/hip_runtime.h>
#include <math.h>

#define MO 48
#define DD 32
#define NN 16

typedef __attribute__((ext_vector_type(2))) float v2f;
typedef __attribute__((ext_vector_type(8))) float v8f;

// workspace layout (float offsets)
#define WS_HM   0        // 16*128  mod hidden (rows 4..15 zero pad)
#define WS_HC   2048     // 32*128  comb hidden
#define WS_LM   6144     // 16*2304 mod logits [cfg][r]
#define WS_LC   43008    // 32*4608 comb logits [cfg][r]
#define WS_MODC 190464   // 4*48    mod attn column 0 [cfg][p]
#define WS_TCMB 190656   // 32*48*96 comb attn [cfg][p][q]

__device__ __forceinline__ float gelu_exact(float x) {
    return 0.5f * x * (1.0f + erff(x * 0.70710678118654752440f));
}

// ---------------- Kernel 1: hidden layers for 16(mod-padded)+32(comb) configs
__global__ void hidden_kernel(const float* __restrict__ mw1, const float* __restrict__ mb1,
                              const float* __restrict__ mw2, const float* __restrict__ mb2,
                              const float* __restrict__ cw1, const float* __restrict__ cb1,
                              const float* __restrict__ cw2, const float* __restrict__ cb2,
                              float* __restrict__ ws) {
    __shared__ float h1[128];
    int t = threadIdx.x;
    int blk = blockIdx.x;
    float x[4];
    int nx;
    const float *W1, *B1, *W2, *B2;
    float* Hout;
    bool valid = true;
    if (blk < 16) {                       // mod config: cfg = c*2 + s
        int cfg = blk, c = cfg >> 1, s = cfg & 1;
        x[0] = (float)c / 48.0f; x[1] = (s == 0) ? 1.0f : 0.0f; x[2] = (s == 1) ? 1.0f : 0.0f; x[3] = 0.0f;
        nx = 3; W1 = mw1; B1 = mb1; W2 = mw2; B2 = mb2;
        Hout = ws + WS_HM + cfg * 128;
        valid = (cfg < 4);
    } else {                              // comb config: cfg = (l*4+r)*2 + s
        int cfg = blk - 16, l = cfg >> 3, r = (cfg >> 1) & 3, s = cfg & 1;
        x[0] = (float)l / 48.0f; x[1] = (float)r / 48.0f;
        x[2] = (s == 0) ? 1.0f : 0.0f; x[3] = (s == 1) ? 1.0f : 0.0f;
        nx = 4; W1 = cw1; B1 = cb1; W2 = cw2; B2 = cb2;
        Hout = ws + WS_HC + cfg * 128;
    }
    float acc = B1[t];
    for (int j = 0; j < nx; ++j) acc += W1[t * nx + j] * x[j];
    h1[t] = gelu_exact(acc);
    __syncthreads();
    float acc2 = B2[t];
    for (int j = 0; j < 128; ++j) acc2 += W2[t * 128 + j] * h1[j];
    Hout[t] = valid ? gelu_exact(acc2) : 0.0f;
}

// ---------------- Kernel 2: WMMA f32 16x16x4 GEMM: logits = W3 @ H^T + b3
// jobs 0..143: mod (2304x16 rows x 1 cfg tile); 144..719: comb (4608 rows x 2 cfg tiles)
__global__ __launch_bounds__(32) void wmma_logits_kernel(const float* __restrict__ mw3, const float* __restrict__ mb3,
                                                         const float* __restrict__ cw3, const float* __restrict__ cb3,
                                                         float* __restrict__ ws) {
    int j = blockIdx.x;
    const float *W3, *H, *B3;
    float* L;
    int r0, n0, stride;
    if (j < 144) {
        W3 = mw3; H = ws + WS_HM; B3 = mb3; L = ws + WS_LM;
        r0 = j * 16; n0 = 0; stride = 2304;
    } else {
        j -= 144;
        int rt = j >> 1, nt = j & 1;
        W3 = cw3; H = ws + WS_HC; B3 = cb3; L = ws + WS_LC;
        r0 = rt * 16; n0 = nt * 16; stride = 4608;
    }
    int lane = threadIdx.x;
    int half = lane >> 4;      // K split: lanes 0-15 hold K,K+1; lanes 16-31 hold K+2,K+3
    int lid  = lane & 15;      // M (A) / N (B) index within half
    v8f acc = {0.f, 0.f, 0.f, 0.f, 0.f, 0.f, 0.f, 0.f};
    const float* wrow = W3 + (size_t)(r0 + lid) * 128 + 2 * half;
    const float* hrow = H  + (size_t)(n0 + lid) * 128 + 2 * half;
    for (int k0 = 0; k0 < 128; k0 += 4) {
        v2f a; a.x = wrow[k0]; a.y = wrow[k0 + 1];   // A: 16x4 (M in lanes, K in VGPRs)
        v2f b; b.x = hrow[k0]; b.y = hrow[k0 + 1];   // B: 4x16 (N in lanes, K in VGPRs), B[k][n]=H[n][k]
        acc = __builtin_amdgcn_wmma_f32_16x16x4_f32(false, a, false, b, (short)0, acc,
                                                    false, false);
    }
    int cfg = n0 + lid;
#pragma unroll
    for (int m = 0; m < 8; ++m) {        // C/D: VGPR m -> rows r0+m (lanes<16) / r0+m+8
        int r = r0 + m + 8 * half;
        L[(size_t)cfg * stride + r] = acc[m] + B3[r];
    }
}

// ---------------- Kernel 3: row softmax. mod rows -> column-0 only; comb rows -> full 96.
__global__ void softmax_kernel(float* __restrict__ ws) {
    __shared__ float red[128];
    int t = threadIdx.x;
    int bid = blockIdx.x;
    const float* src;
    float* dst;
    int W;
    bool col0;
    if (bid < 192) {  // 4 cfg * 48 rows
        int cfg = bid / 48, p = bid % 48;
        W = 48; col0 = true;
        src = ws + WS_LM + cfg * 2304 + p * 48;
        dst = ws + WS_MODC + cfg * 48 + p;
    } else {          // 32 cfg * 48 rows
        int i = bid - 192;
        int cfg = i / 48, p = i % 48;
        W = 96; col0 = false;
        src = ws + WS_LC + (size_t)cfg * 4608 + p * 96;
        dst = ws + WS_TCMB + (size_t)cfg * 4608 + p * 96;
    }
    float v = (t < W) ? src[t] : -INFINITY;
    red[t] = v;
    __syncthreads();
    for (int off = 64; off > 0; off >>= 1) {
        if (t < off) red[t] = fmaxf(red[t], red[t + off]);
        __syncthreads();
    }
    float mx = red[0];
    __syncthreads();
    float e = (t < W) ? expf(v - mx) : 0.0f;
    red[t] = e;
    __syncthreads();
    for (int off = 64; off > 0; off >>= 1) {
        if (t < off) red[t] += red[t + off];
        __syncthreads();
    }
    float s = red[0];
    if (col0) {
        if (t == 0) dst[0] = e / s;
    } else if (t < W) {
        dst[t] = e / s;
    }
}

// ---------------- Kernel 4: per-batch rank-2 reconstruction + projection
struct Term { int uKind; int uCfg; int vIdx; float vScl; float cnt; };
// uKind: 0 = zero buffer, 1 = e0 (prim), 2 = mod attn column uCfg

__device__ Term phase2_term(const int* cats, const int* subs, const int* mask,
                            const int* clv, int base, int n) {
    Term tm; tm.uKind = 0; tm.uCfg = 0; tm.vIdx = 0; tm.vScl = 0.0f; tm.cnt = 0.0f;
    int cat = cats[base + n];
    int m = mask[base + n];
    int s = subs[base + n]; s = s < 0 ? 0 : (s > 1 ? 1 : s);
    if (!m) return tm;
    if (cat == 0) {  // prim
        int aid = subs[base + n] + 1; aid = aid < 0 ? 0 : (aid > 7 ? 7 : aid);
        tm.uKind = 1; tm.vIdx = aid; tm.vScl = 1.0f; tm.cnt = 1.0f;
        return tm;
    }
    if (cat == 1) {  // mod: rank-1 = modAttnCol0[c][s] (x) prim_vec(child)
        int c0 = clv[base + n]; c0 = c0 < 0 ? 0 : (c0 > 15 ? 15 : c0);
        int cprim = (cats[base + c0] == 0 && mask[base + c0]) ? 1 : 0;
        int aid = subs[base + c0] + 1; aid = aid < 0 ? 0 : (aid > 7 ? 7 : aid);
        tm.uKind = 2; tm.uCfg = cprim * 2 + s; tm.vIdx = aid; tm.vScl = (float)cprim;
        float cnt = (float)cprim * (s ? 3.0f : 2.0f);
        tm.cnt = cnt > 48.0f ? 48.0f : cnt;
        return tm;
    }
    return tm;  // comb or other: buffer still zero at phase 2
}

__global__ void finalize_kernel(const int* __restrict__ cats, const int* __restrict__ subs,
                                const int* __restrict__ mask, const int* __restrict__ clv,
                                const int* __restrict__ crv, const float* __restrict__ E,
                                const float* __restrict__ ws, float* __restrict__ out) {
    __shared__ float uA[48], uB[48], u1[48], u2[48], v1[32], v2[32], ed1[8], ed2[8];
    __shared__ int sComb, sCfg;
    __shared__ float sCnt;
    __shared__ Term sT[2];
    int b = blockIdx.x;
    int t = threadIdx.x;
    int base = b * NN;
    if (t == 0) {
        int cat0 = cats[base], m0 = mask[base];
        int s0 = subs[base]; s0 = s0 < 0 ? 0 : (s0 > 1 ? 1 : s0);
        if (cat0 == 2 && m0) {
            int l0 = clv[base]; l0 = l0 < 0 ? 0 : (l0 > 15 ? 15 : l0);
            int r0 = crv[base]; r0 = r0 < 0 ? 0 : (r0 > 15 ? 15 : r0);
            Term tl = phase2_term(cats, subs, mask, clv, base, l0);
            Term tr = phase2_term(cats, subs, mask, clv, base, r0);
            sT[0] = tl; sT[1] = tr; sComb = 1;
            int iL = (int)tl.cnt, iR = (int)tr.cnt;  // in {0..3}
            sCfg = (iL * 4 + iR) * 2 + s0;
            float c = tl.cnt + tr.cnt;
            sCnt = c > 48.0f ? 48.0f : c;
        } else {
            Term t0 = phase2_term(cats, subs, mask, clv, base, 0);
            Term z; z.uKind = 0; z.uCfg = 0; z.vIdx = 0; z.vScl = 0.0f; z.cnt = 0.0f;
            sT[0] = t0; sT[1] = z; sComb = 0; sCfg = 0; sCnt = t0.cnt;
        }
    }
    __syncthreads();
    const float* modC = ws + WS_MODC;
    if (t < 48) {
        Term tl = sT[0], tr = sT[1];
        uA[t] = (tl.uKind == 1) ? (t == 0 ? 1.0f : 0.0f)
              : (tl.uKind == 2) ? modC[tl.uCfg * 48 + t] : 0.0f;
        uB[t] = (tr.uKind == 1) ? (t == 0 ? 1.0f : 0.0f)
              : (tr.uKind == 2) ? modC[tr.uCfg * 48 + t] : 0.0f;
    }
    __syncthreads();
    if (t < 48) {
        if (sComb) {
            const float* A = ws + WS_TCMB + (size_t)sCfg * 4608 + t * 96;
            float a1 = 0.0f, a2 = 0.0f;
            for (int q = 0; q < 48; ++q) {
                a1 += A[q] * uA[q];
                a2 += A[48 + q] * uB[q];
            }
            u1[t] = a1; u2[t] = a2;
        } else {
            u1[t] = uA[t]; u2[t] = uB[t];
        }
    }
    if (t < 32) {
        v1[t] = sT[0].vScl * E[sT[0].vIdx * 32 + t];
        v2[t] = sT[1].vScl * E[sT[1].vIdx * 32 + t];
    }
    __syncthreads();
    if (t < 8) {
        float a = 0.0f, c = 0.0f;
        for (int d = 0; d < 32; ++d) {
            a += E[t * 32 + d] * v1[d];
            c += E[t * 32 + d] * v2[d];
        }
        ed1[t] = a; ed2[t] = c;
    }
    __syncthreads();
    for (int i = t; i < MO * 8; i += 64) {
        int p = i >> 3, a = i & 7;
        float val = u1[p] * ed1[a] + u2[p] * ed2[a];
        if (a == 0 && (float)p >= sCnt) val += 8.0f;  // STOP bias
        out[(size_t)b * (MO * 8) + i] = val;
    }
}

extern "C" void kernel_launch(void* const* d_in, const int* in_sizes, int n_in,
                              void* d_out, int out_size, void* d_ws, size_t ws_size,
                              hipStream_t stream) {
    const int* cats = (const int*)d_in[0];
    const int* subs = (const int*)d_in[1];
    const int* mask = (const int*)d_in[2];
    const int* clv  = (const int*)d_in[3];
    const int* crv  = (const int*)d_in[4];
    const float* E   = (const float*)d_in[5];
    const float* mw1 = (const float*)d_in[6];
    const float* mb1 = (const float*)d_in[7];
    const float* mw2 = (const float*)d_in[8];
    const float* mb2 = (const float*)d_in[9];
    const float* mw3 = (const float*)d_in[10];
    const float* mb3 = (const float*)d_in[11];
    const float* cw1 = (const float*)d_in[12];
    const float* cb1 = (const float*)d_in[13];
    const float* cw2 = (const float*)d_in[14];
    const float* cb2 = (const float*)d_in[15];
    const float* cw3 = (const float*)d_in[16];
    const float* cb3 = (const float*)d_in[17];
    float* ws  = (float*)d_ws;
    float* out = (float*)d_out;
    int B = in_sizes[0] / NN;

    hipLaunchKernelGGL(hidden_kernel, dim3(48), dim3(128), 0, stream,
                       mw1, mb1, mw2, mb2, cw1, cb1, cw2, cb2, ws);
    hipLaunchKernelGGL(wmma_logits_kernel, dim3(720), dim3(32), 0, stream,
                       mw3, mb3, cw3, cb3, ws);
    hipLaunchKernelGGL(softmax_kernel, dim3(1728), dim3(128), 0, stream, ws);
    hipLaunchKernelGGL(finalize_kernel, dim3(B), dim3(64), 0, stream,
                       cats, subs, mask, clv, crv, E, ws, out);
}